// Actor_46566035423403
// MI455X (gfx1250) — compile-verified
//
#include <hip/hip_runtime.h>
#include <hip/hip_bf16.h>
#include <math.h>

// Sizes from the reference
#define IN_SIZE 65536
#define HIDDEN  128
#define D_STEPS 64
#define N_ACT   512
#define GATES   (4 * HIDDEN)   // 512

// LDS layout for the LSTM kernel (single dynamic region, base offset known):
// w_hh staged with padded row stride to avoid bank conflicts.
#define WHH_ROW_F   132                       // 128 floats + 4 pad (16 B)
#define WHH_ROW_B   (WHH_ROW_F * 4)           // 528 bytes
#define WHH_LDS_B   (GATES * WHH_ROW_B)       // 270336 B
#define LSTM_LDS_B  (WHH_LDS_B + (HIDDEN + HIDDEN + GATES) * 4)  // +3 KB

typedef __attribute__((ext_vector_type(2))) float v2f;
typedef __attribute__((ext_vector_type(8))) float v8f;

// ---------------------------------------------------------------------------
// Kernel 1: ihx = w_ih @ x + b_ih + b_hh
// 512 workgroups (one per output row) x 256 threads, float4 streaming loads.
// Pure HBM-bandwidth phase: 128 MB once-through at ~23.3 TB/s (~5.5 us floor).
// ---------------------------------------------------------------------------
__global__ __launch_bounds__(256) void gemv_ihx_kernel(
    const float* __restrict__ w_ih, const float* __restrict__ x,
    const float* __restrict__ b_ih, const float* __restrict__ b_hh,
    float* __restrict__ ihx) {
  __shared__ float red[256];
  const int row = blockIdx.x;
  const int t   = threadIdx.x;
  const float4* __restrict__ w4 = (const float4*)(w_ih + (size_t)row * IN_SIZE);
  const float4* __restrict__ x4 = (const float4*)x;

  float sum = 0.0f;
  // 16384 float4 per row, 256 threads -> 64 iterations, fully coalesced b128.
  for (int i = t; i < IN_SIZE / 4; i += 256) {
    // speculative prefetch of the next strip (gfx1250 global_prefetch_b8;
    // OOB speculative prefetches are silently dropped)
    __builtin_prefetch(&w4[i + 256], 0, 0);
    float4 w = w4[i];
    float4 xv = x4[i];
    sum += w.x * xv.x + w.y * xv.y + w.z * xv.z + w.w * xv.w;
  }
  red[t] = sum;
  __syncthreads();
  for (int s = 128; s > 0; s >>= 1) {
    if (t < s) red[t] += red[t + s];
    __syncthreads();
  }
  if (t == 0) ihx[row] = red[0] + b_ih[row] + b_hh[row];
}

// ---------------------------------------------------------------------------
// Kernel 2: sequential LSTM recurrence (64 steps).
// One workgroup, 512 threads (16 wave32s). All of w_hh (256 KB) is staged
// into LDS once via GLOBAL_LOAD_ASYNC_TO_LDS_B128 (ASYNCcnt path) — 320 KB
// LDS/WGP on CDNA5 makes this possible. Every serial step then runs its
// 128-MAC row dot out of LDS instead of L2 (w_hh >> 64 KB WGP$).
// Rows padded to 132 floats so lanes of a wave hit distinct LDS banks.
// ---------------------------------------------------------------------------
__global__ __launch_bounds__(512) void lstm_seq_kernel(
    const float* __restrict__ w_hh, const float* __restrict__ ihx,
    float* __restrict__ h_all) {
  extern __shared__ char smem[];
  float* whh_lds = (float*)smem;                     // [512][132]
  float* h       = (float*)(smem + WHH_LDS_B);       // [128]
  float* c       = h + HIDDEN;                       // [128]
  float* gates   = c + HIDDEN;                       // [512]

  const int t = threadIdx.x;  // 0..511, one w_hh row per thread

  // --- async-stage w_hh row t into LDS (32 x b128 per thread) -------------
  // Low 32 bits of the flat shared-pointer are the wave-relative LDS byte
  // address (ISA 10.2 aperture rules), which is what the async op's VDST
  // operand expects.
  const unsigned lds_row  = (unsigned)(unsigned long long)whh_lds + t * WHH_ROW_B;
  const unsigned gbl_row  = (unsigned)t * (HIDDEN * 4);  // byte offset in w_hh
#pragma unroll
  for (int j = 0; j < (HIDDEN * 4) / 16; ++j) {          // 32 x 16 B = 512 B
    asm volatile(
        "global_load_async_to_lds_b128 %0, %1, %2"
        :
        : "v"(lds_row + j * 16), "v"(gbl_row + j * 16), "s"(w_hh)
        : "memory");
  }

  if (t < HIDDEN) { h[t] = 0.0f; c[t] = 0.0f; }
  const float ihx_t = ihx[t];

  asm volatile("s_wait_asynccnt 0" ::: "memory");  // drain ASYNCcnt
  __syncthreads();

  const float4* __restrict__ wrow = (const float4*)(whh_lds + t * WHH_ROW_F);

  for (int step = 0; step < D_STEPS; ++step) {
    // dot(w_hh[t,:], h) : 128 MACs, both operands in LDS
    float dot = 0.0f;
#pragma unroll 8
    for (int k = 0; k < HIDDEN / 4; ++k) {
      float4 w = wrow[k];
      dot += w.x * h[4 * k + 0] + w.y * h[4 * k + 1] +
             w.z * h[4 * k + 2] + w.w * h[4 * k + 3];
    }
    gates[t] = ihx_t + dot;
    __syncthreads();

    if (t < HIDDEN) {
      // torch LSTMCell gate order: i, f, g, o
      float gi = gates[t];
      float gf = gates[HIDDEN + t];
      float gg = gates[2 * HIDDEN + t];
      float go = gates[3 * HIDDEN + t];
      float si = 1.0f / (1.0f + __expf(-gi));
      float sf = 1.0f / (1.0f + __expf(-gf));
      float so = 1.0f / (1.0f + __expf(-go));
      float cn = sf * c[t] + si * tanhf(gg);
      c[t] = cn;
      float hn = so * tanhf(cn);
      h[t] = hn;
      h_all[step * HIDDEN + t] = hn;  // feed the WMMA GEMM
    }
    __syncthreads();
  }
}

// ---------------------------------------------------------------------------
// Kernel 3: logits(64x512) = h_all(64x128) @ w_lin^T(128x512) + b_lin
// Full-precision fp32 WMMA: V_WMMA_F32_16X16X4_F32, one 16x16 tile per wave,
// 32 k-steps (K=128). Grid: 4 Mtiles x 32 Ntiles = 128 single-wave blocks.
//
// Layouts (ISA 7.12.2):
//   A 16x4 f32 : 2 VGPRs; lanes 0-15 -> M=lane, {v0,v1}={K0,K1};
//                lanes 16-31 -> M=lane-16, {v0,v1}={K2,K3}
//   B 4x16 f32 : mirrored, lanes = N, lane-half selects K pair
//   D 16x16 f32: 8 VGPRs; vgpr r -> M = r + 8*(lane>=16), N = lane&15
// ---------------------------------------------------------------------------
__global__ __launch_bounds__(32) void logits_wmma_kernel(
    const float* __restrict__ h_all, const float* __restrict__ w_lin,
    const float* __restrict__ b_lin, float* __restrict__ out) {
  const int tile  = blockIdx.x;        // 0..127
  const int m0    = (tile >> 5) << 4;  // 4 M-tiles of 16 (steps)
  const int n0    = (tile & 31) << 4;  // 32 N-tiles of 16 (actions)
  const int lane  = threadIdx.x;       // 0..31 (wave32, EXEC all ones)
  const int row   = lane & 15;
  const int khalf = (lane >> 4) << 1;  // 0 for lanes 0-15, 2 for lanes 16-31

  v8f acc = {};
#pragma unroll
  for (int kk = 0; kk < HIDDEN / 4; ++kk) {
    const int k0 = kk * 4 + khalf;
    v2f a, b;
    // A: h_all[step = m0+row][k]
    a.x = h_all[(m0 + row) * HIDDEN + k0 + 0];
    a.y = h_all[(m0 + row) * HIDDEN + k0 + 1];
    // B: w_lin^T[k][n = n0+row] = w_lin[n][k]  (w_lin is N_ACT x HIDDEN)
    b.x = w_lin[(n0 + row) * HIDDEN + k0 + 0];
    b.y = w_lin[(n0 + row) * HIDDEN + k0 + 1];
    // D = A x B + C  (fp32, round-to-nearest-even)
    acc = __builtin_amdgcn_wmma_f32_16x16x4_f32(
        /*neg_a=*/false, a, /*neg_b=*/false, b,
        /*c_mod=*/(short)0, acc, /*reuse_a=*/false, /*reuse_b=*/false);
  }

  const int mbase = m0 + ((lane >> 4) << 3);  // +8 for upper half-wave
  const int n     = n0 + row;
  const float bl  = b_lin[n];
#pragma unroll
  for (int r = 0; r < 8; ++r) {
    out[(mbase + r) * N_ACT + n] = acc[r] + bl;
  }
}

// ---------------------------------------------------------------------------
// Launch: inputs in setup_inputs() order:
//   0:x  1:w_ih  2:w_hh  3:b_ih  4:b_hh  5:w_lin  6:b_lin
// Output: (64, 512) float32.
// Workspace: ihx (512 f32) | h_all (64*128 f32)
// ---------------------------------------------------------------------------
extern "C" void kernel_launch(void* const* d_in, const int* in_sizes, int n_in,
                              void* d_out, int out_size, void* d_ws, size_t ws_size,
                              hipStream_t stream) {
  const float* x     = (const float*)d_in[0];
  const float* w_ih  = (const float*)d_in[1];
  const float* w_hh  = (const float*)d_in[2];
  const float* b_ih  = (const float*)d_in[3];
  const float* b_hh  = (const float*)d_in[4];
  const float* w_lin = (const float*)d_in[5];
  const float* b_lin = (const float*)d_in[6];
  float* out = (float*)d_out;

  float* ihx   = (float*)d_ws;          // 512 floats
  float* h_all = ihx + GATES;           // 64*128 floats

  // Allow > default dynamic LDS (267 KB of the 320 KB/WGP CDNA5 budget).
  (void)hipFuncSetAttribute((const void*)lstm_seq_kernel,
                            hipFuncAttributeMaxDynamicSharedMemorySize,
                            LSTM_LDS_B);

  gemv_ihx_kernel<<<GATES, 256, 0, stream>>>(w_ih, x, b_ih, b_hh, ihx);
  lstm_seq_kernel<<<1, 512, LSTM_LDS_B, stream>>>(w_hh, ihx, h_all);
  logits_wmma_kernel<<<(D_STEPS / 16) * (N_ACT / 16), 32, 0, stream>>>(
      h_all, w_lin, b_lin, out);
}